// GuassModel_cuda_51814485459548
// MI455X (gfx1250) — compile-verified
//
#include <hip/hip_runtime.h>
#include <hip/hip_bf16.h>

typedef __attribute__((ext_vector_type(2))) float v2f;
typedef __attribute__((ext_vector_type(4))) float v4f;
typedef __attribute__((ext_vector_type(8))) float v8f;

#define DIM0 256
#define DIM1 256
#define DIM2 256
#define NGAUSS 65536
#define WAVES_PER_BLOCK 8

// ---------------------------------------------------------------------------
// Kernel 1: zero-fill output (we accumulate with atomics, and graph replays
// re-run the whole launch, so the clear must happen every call).
// ---------------------------------------------------------------------------
__global__ __launch_bounds__(256) void zero_f4_kernel(float4* __restrict__ out, int n4) {
    int i = blockIdx.x * blockDim.x + threadIdx.x;
    if (i < n4) out[i] = make_float4(0.f, 0.f, 0.f, 0.f);
}

// ---------------------------------------------------------------------------
// Kernel 2: one wave32 per Gaussian.
//   q[m,n] = f[m] + g[n] + h[m]*dz[n]   (m = ox*7+oy in 0..48, n = oz in 0..6)
// computed as D = A(16x4) x B(4x16) with V_WMMA_F32_16X16X4_F32:
//   A rows m:  K0=f[m], K1=1, K2=h[m], K3=0
//   B cols n:  K0=1,    K1=g[n], K2=dz[n], K3=0
// 4 chunks of 16 rows cover m=0..63 (padded), columns n=0..15 (padded).
// ---------------------------------------------------------------------------
__global__ __launch_bounds__(256) void splat_wmma_kernel(
        const float* __restrict__ paras,    // (10, N) row-major
        const float* __restrict__ dist_p,   // scalar
        const float* __restrict__ thr_p,    // scalar
        float* __restrict__ out)            // 256^3
{
    // per-wave LDS tables: f[64], h[64], s[64] (=dx^2+dy^2), g[16], dz[16], t[16] (=dz^2)
    // row size 240 floats = 960 B (16B aligned) so b128 LDS loads stay aligned.
    __shared__ float smem[WAVES_PER_BLOCK][240];
    const int lane = threadIdx.x & 31;
    const int wave = threadIdx.x >> 5;
    const int g    = blockIdx.x * WAVES_PER_BLOCK + wave;
    float* S = smem[wave];

    // Wave-uniform parameter loads (L2-cached; redundant across lanes is fine).
    const float px  = paras[0 * NGAUSS + g];
    const float py  = paras[1 * NGAUSS + g];
    const float pz  = paras[2 * NGAUSS + g];
    const float val = paras[3 * NGAUSS + g];
    const float rx  = paras[4 * NGAUSS + g];
    const float rxy = paras[5 * NGAUSS + g];
    const float rxz = paras[6 * NGAUSS + g];
    const float ry  = paras[7 * NGAUSS + g];
    const float ryz = paras[8 * NGAUSS + g];
    const float rz  = paras[9 * NGAUSS + g];
    const float dist     = dist_p[0];
    const float thr      = thr_p[0];
    const float dist2max = dist * dist;

    const int   cx  = (int)floorf(px) - 3;
    const int   cy  = (int)floorf(py) - 3;
    const int   cz  = (int)floorf(pz) - 3;
    const float fxp = px - (float)cx;   // voxel-relative center, in [3,4)
    const float fyp = py - (float)cy;
    const float fzp = pz - (float)cz;
    const float irx = 1.0f / rx, iry = 1.0f / ry, irz = 1.0f / rz;

    // Fill per-m tables (m = lane and m = lane+32); pad m=49..63 with zeros.
    #pragma unroll
    for (int b = 0; b < 2; ++b) {
        const int m = lane + 32 * b;
        float f = 0.f, h = 0.f, s = 0.f;
        if (m < 49) {
            const int ox = m / 7;
            const int oy = m - ox * 7;
            const float dx = (float)ox - fxp;
            const float dy = (float)oy - fyp;
            const float ax = dx * irx;
            const float ay = dy * iry;
            f = ax * ax + ay * ay + rxy * dx * dy;
            h = rxz * dx + ryz * dy;
            s = dx * dx + dy * dy;
        }
        S[m] = f; S[64 + m] = h; S[128 + m] = s;
    }
    // Per-n tables; pad n=7..15 with zeros.
    if (lane < 16) {
        float gq = 0.f, dzv = 0.f, t = 0.f;
        if (lane < 7) {
            const float dz = (float)lane - fzp;
            const float az = dz * irz;
            gq = az * az; dzv = dz; t = dz * dz;
        }
        S[192 + lane] = gq; S[208 + lane] = dzv; S[224 + lane] = t;
    }
    __syncthreads();   // make cross-lane LDS tables visible

    const int  n      = lane & 15;
    const bool hiHalf = lane >= 16;
    const bool colOK  = n < 7;
    // hoisted loop invariants for the scatter phase
    const float t_n   = S[224 + n];          // dz^2 for this column
    const int   vz    = cz + n;
    const bool  zOK   = colOK && (vz >= 0) && (vz < DIM2);

    #pragma unroll
    for (int c = 0; c < 4; ++c) {
        const int mb = c * 16 + (lane & 15);
        // A (16x4): v0 = {K0 | K2}, v1 = {K1 | K3}  (low lanes | high lanes)
        v2f A;
        A.x = hiHalf ? S[64 + mb] : S[mb];      // K2=h[m]  | K0=f[m]
        A.y = hiHalf ? 0.0f       : 1.0f;       // K3=0     | K1=1
        // B (4x16): v0 = {K0 row | K2 row}, v1 = {K1 row | K3 row}
        v2f B;
        B.x = hiHalf ? S[208 + n] : 1.0f;       // K2=dz[n] | K0=1
        B.y = hiHalf ? 0.0f       : S[192 + n]; // K3=0     | K1=g[n]
        v8f q = {};
        q = __builtin_amdgcn_wmma_f32_16x16x4_f32(
                /*neg_a=*/false, A, /*neg_b=*/false, B,
                /*c_mod=*/(short)0, q, /*reuse_a=*/false, /*reuse_b=*/false);

        if (zOK) {
            const int mbase = c * 16 + (hiHalf ? 8 : 0);   // D layout: M=j / M=j+8
            // s[mbase..mbase+7] are 8 consecutive, 16B-aligned floats: 2x ds_load_b128
            const v4f* sp = (const v4f*)&S[128 + mbase];
            const v4f s_lo = sp[0];
            const v4f s_hi = sp[1];

            #pragma unroll
            for (int j = 0; j < 8; ++j) {
                const int m = mbase + j;
                if (m < 49) {
                    const int ox = m / 7;
                    const int oy = m - ox * 7;
                    const int vx = cx + ox, vy = cy + oy;
                    const float sm    = (j < 4) ? s_lo[j & 3] : s_hi[j & 3];
                    const float dist2 = sm + t_n;
                    if ((vx >= 0) && (vx < DIM0) && (vy >= 0) && (vy < DIM1) &&
                        (dist2 <= dist2max)) {
                        const float w = __expf(-0.5f * q[j]);
                        if (w > thr) {
                            unsafeAtomicAdd(&out[(vx * DIM1 + vy) * DIM2 + vz], val * w);
                        }
                    }
                }
            }
        }
    }
}

// ---------------------------------------------------------------------------
extern "C" void kernel_launch(void* const* d_in, const int* in_sizes, int n_in,
                              void* d_out, int out_size, void* d_ws, size_t ws_size,
                              hipStream_t stream) {
    const float* paras  = (const float*)d_in[0];   // (10, 65536)
    const float* dist_p = (const float*)d_in[1];   // scalar
    const float* thr_p  = (const float*)d_in[2];   // scalar
    float* out = (float*)d_out;                    // 256^3 floats

    const int n4 = (DIM0 * DIM1 * DIM2) / 4;       // 4,194,304 float4
    zero_f4_kernel<<<n4 / 256, 256, 0, stream>>>((float4*)out, n4);

    const int blocks = NGAUSS / WAVES_PER_BLOCK;   // 8192 blocks x 8 waves
    splat_wmma_kernel<<<blocks, 256, 0, stream>>>(paras, dist_p, thr_p, out);
}